// SimulateBatchedFullAttn_78572131713231
// MI455X (gfx1250) — compile-verified
//
#include <hip/hip_runtime.h>

// ---------------------------------------------------------------------------
// MI455X (gfx1250) implementation.
//
//   Pass 1: one block per segment -> seg_abs[s] = sum_{rows in seg} sum_n |y_n|
//           (no atomics, no searchsorted; contiguous reads of x; 134 MB HBM;
//           default-RT loads leave x resident in the 192 MB L2)
//   Pass 2: WMMA f32 16x16x4 recomputes y = x @ W^T in 16-row tiles.
//           x re-reads are non-temporal (last use, served from L2); the
//           134 MB output stream is stored non-temporally so it cannot evict
//           the not-yet-read portion of x from L2. D tiles are transposed
//           through per-wave-private LDS, each lane then owns one row:
//           LDS binary-search of the segment, scale, coalesced b128 store.
// ---------------------------------------------------------------------------

typedef __attribute__((ext_vector_type(2))) float v2f;
typedef __attribute__((ext_vector_type(4))) float v4f;
typedef __attribute__((ext_vector_type(8))) float v8f;

#define NSEG 4096

// ----------------------------- Pass 1 --------------------------------------
__global__ __launch_bounds__(256) void seg_sum_kernel(
    const float* __restrict__ x, const int* __restrict__ slices,
    const float* __restrict__ W, float* __restrict__ seg_abs) {
  __shared__ float red[8];
  const int s  = blockIdx.x;
  const int r0 = slices[s];
  const int r1 = slices[s + 1];

  float w[16];
#pragma unroll
  for (int i = 0; i < 16; ++i) w[i] = W[i];

  float acc = 0.f;
  for (int r = r0 + (int)threadIdx.x; r < r1; r += 256) {
    const v4f xv = *(const v4f*)(x + (size_t)r * 4);
#pragma unroll
    for (int n = 0; n < 4; ++n) {
      const float y = xv.x * w[n * 4 + 0] + xv.y * w[n * 4 + 1] +
                      xv.z * w[n * 4 + 2] + xv.w * w[n * 4 + 3];
      acc += fabsf(y);
    }
  }
  // wave32 reduction
#pragma unroll
  for (int off = 16; off > 0; off >>= 1) acc += __shfl_down(acc, off, 32);
  if ((threadIdx.x & 31) == 0) red[threadIdx.x >> 5] = acc;
  __syncthreads();
  if (threadIdx.x == 0) {
    float t = 0.f;
#pragma unroll
    for (int j = 0; j < 8; ++j) t += red[j];
    seg_abs[s] = t;
  }
}

// ----------------------------- Pass 2 --------------------------------------
// A layout  (16x4 f32, 2 VGPRs): lane l, vgpr v -> M = l%16, K = v + 2*(l/16)
// B layout  (4x16 f32, 2 VGPRs): lane l, vgpr v -> N = l%16, K = v + 2*(l/16)
// D layout  (16x16 f32, 8 VGPRs): lane l, vgpr g -> M = g + 8*(l/16), N = l%16
__global__ __launch_bounds__(256) void wmma_norm_kernel(
    const float* __restrict__ x, const int* __restrict__ slices,
    const float* __restrict__ W, const float* __restrict__ seg_abs,
    float* __restrict__ out, int nchunks) {
  __shared__ int s_sl[NSEG + 1];  // 16388 B; source stays L2-resident
  __shared__ v4f s_y[8][32];      // per-wave private 32-row transpose tile

  for (int i = threadIdx.x; i <= NSEG; i += 256) s_sl[i] = slices[i];
  __syncthreads();

  const int lane = threadIdx.x & 31;
  const int wave = threadIdx.x >> 5;
  const int n    = lane & 15;  // N for B/D, M for A
  const int hi   = lane >> 4;  // lane-half selector

  // B = W^T in columns 0..3, zero elsewhere: B[k][n] = W[n*4 + k]
  v2f bmat;
  if (n < 4) {
    bmat.x = W[n * 4 + 2 * hi + 0];
    bmat.y = W[n * 4 + 2 * hi + 1];
  } else {
    bmat.x = 0.f;
    bmat.y = 0.f;
  }

  for (int chunk = blockIdx.x; chunk < nchunks; chunk += gridDim.x) {
    const int waveBase = chunk * 256 + wave * 32;

    // WGP-scope prefetch of the next grid-stride chunk (global_prefetch_b8):
    // pull it from L2 (resident after pass 1) into the near cache.
    const int nextChunk = chunk + gridDim.x;
    if (nextChunk < nchunks)
      __builtin_prefetch(x + (size_t)(nextChunk * 256 + wave * 32 + lane) * 4,
                         0, 3);

    // Issue BOTH A-tile loads before the first WMMA (overlap latency).
    // Non-temporal: this is the last use of x, don't keep it cached.
    const v2f a0 = __builtin_nontemporal_load(
        (const v2f*)(x + (size_t)(waveBase + n) * 4 + 2 * hi));
    const v2f a1 = __builtin_nontemporal_load(
        (const v2f*)(x + (size_t)(waveBase + 16 + n) * 4 + 2 * hi));

    v8f c = {0.f, 0.f, 0.f, 0.f, 0.f, 0.f, 0.f, 0.f};
    // (neg_a, A, neg_b, B, c_mod, C, reuse_a, reuse_b)
    const v8f d0 = __builtin_amdgcn_wmma_f32_16x16x4_f32(
        false, a0, false, bmat, (short)0, c, false, false);
    const v8f d1 = __builtin_amdgcn_wmma_f32_16x16x4_f32(
        false, a1, false, bmat, (short)0, c, false, false);

    // Scatter useful D columns (n<4) into the per-wave row-major tile.
    if (n < 4) {
      float* dst0 = (float*)&s_y[wave][8 * hi];
      float* dst1 = (float*)&s_y[wave][16 + 8 * hi];
#pragma unroll
      for (int g = 0; g < 8; ++g) {
        dst0[g * 4 + n] = d0[g];
        dst1[g * 4 + n] = d1[g];
      }
    }

    // Per-wave LDS RAW: DS ops from one wave are in-order; wait for the
    // writes before the cross-lane reads below.
    asm volatile("s_wait_dscnt 0" ::: "memory");

    // Each lane now owns one full row: segment lookup + scale + b128 store.
    const int row = waveBase + lane;
    v4f yv = s_y[wave][lane];

    int lo = 0, hib = NSEG + 1;  // upper_bound(slices, row) over 4097 entries
    while (lo < hib) {
      const int mid = (lo + hib) >> 1;
      if (s_sl[mid] <= row) lo = mid + 1; else hib = mid;
    }
    int seg = lo - 1;
    seg = seg < 0 ? 0 : (seg > NSEG - 1 ? NSEG - 1 : seg);

    const float inv = 1.0f / seg_abs[seg];  // seg_abs is L2-resident (16 KB)
    yv *= inv;
    // Non-temporal store: the 134 MB output stream must not evict the
    // still-unread portion of x from L2.
    __builtin_nontemporal_store(yv, (v4f*)(out + (size_t)row * 4));

    // WAR guard before next iteration overwrites the per-wave tile.
    asm volatile("s_wait_dscnt 0" ::: "memory");
  }
}

// ----------------------------- Launcher ------------------------------------
extern "C" void kernel_launch(void* const* d_in, const int* in_sizes, int n_in,
                              void* d_out, int out_size, void* d_ws,
                              size_t ws_size, hipStream_t stream) {
  const float* x      = (const float*)d_in[0];
  const int*   slices = (const int*)d_in[1];
  const float* W      = (const float*)d_in[2];
  float*       out    = (float*)d_out;
  float*       segs   = (float*)d_ws;  // 4096 floats; fully overwritten each call

  const int nrows = in_sizes[0] / 4;

  // Pass 1: one block per segment.
  seg_sum_kernel<<<NSEG, 256, 0, stream>>>(x, slices, W, segs);

  // Pass 2: grid-stride over 256-row chunks (amortizes per-block LDS preload
  // of `slices`; 2048 blocks of 8 waves saturate the device).
  const int nchunks = nrows / 256;
  const int grid    = nchunks < 2048 ? nchunks : 2048;
  wmma_norm_kernel<<<grid, 256, 0, stream>>>(x, slices, W, segs, out, nchunks);
}